// SelfAttentionBlock_61340722921658
// MI455X (gfx1250) — compile-verified
//
#include <hip/hip_runtime.h>
#include <hip/hip_bf16.h>

typedef float v2f __attribute__((ext_vector_type(2)));
typedef float v8f __attribute__((ext_vector_type(8)));

// ---------------------------------------------------------------------------
// Problem constants (from reference): V=8192, MAXN=16, N=8, C=16, B=8
// ---------------------------------------------------------------------------
#define NV    8192
#define MAXN  16
#define GN    8      // group order N
#define CH    16     // channels C
#define NB    8      // basis count B
#define WAVES 4      // waves (vertices) per block

// workspace layout (floats):
//   [0      .. 1023]  WKt [cp=128][i=8]          W_K^T flattened
//   [1024   .. 3071]  WQpk[cp2=64][m=16][e=2]    W_Q packed for WMMA B operand
//   [3072   .. 9215]  Ppk [c=16][pp=4][m=48][e=2] value-kernel matrices packed
#define WS_WKT   0
#define WS_WQPK  1024
#define WS_PPK   3072
#define WS_TOTAL 9216

// ---------------------------------------------------------------------------
// Prep kernel: fold learned coefficients into dense small matrices, packed in
// the exact per-lane layout the WMMA B operand wants (float2 per lane).
// ---------------------------------------------------------------------------
__global__ void prep_kernel(const float* __restrict__ basis,  // [B][8][8]
                            const float* __restrict__ qc,     // [C][B]
                            const float* __restrict__ kc,     // [C][B]
                            const float* __restrict__ vb0,    // [B][1][8][8]
                            const float* __restrict__ vb1,    // [B][2][8][8]
                            const float* __restrict__ vb2,    // [B][3][8][8]
                            const float* __restrict__ vp0,    // [C][B]
                            const float* __restrict__ vp1,    // [C][B]
                            const float* __restrict__ vp2,    // [C][B]
                            float* __restrict__ ws)
{
    const int stride = blockDim.x * gridDim.x;
    const int t0 = blockIdx.x * blockDim.x + threadIdx.x;

    // W_K^T: WKt[cp][i] = sum_b kc[c][b]*basis[b][i][p], cp = c*8+p
    for (int idx = t0; idx < 1024; idx += stride) {
        int i = idx & 7, cp = idx >> 3, p = cp & 7, c = cp >> 3;
        float s = 0.f;
        for (int b = 0; b < NB; b++) s += kc[c*NB + b] * basis[b*64 + i*8 + p];
        ws[WS_WKT + idx] = s;
    }
    // W_Q packed: WQpk[cp2][m][e] = W_Q[c][i=m][p] with cp=2*cp2+e (m>=8 -> 0 pad)
    for (int idx = t0; idx < 2048; idx += stride) {
        int e = idx & 1, m = (idx >> 1) & 15, cp2 = idx >> 5;
        int cp = cp2*2 + e, p = cp & 7, c = cp >> 3, i = m;
        float s = 0.f;
        if (m < 8)
            for (int b = 0; b < NB; b++) s += qc[c*NB + b] * basis[b*64 + i*8 + p];
        ws[WS_WQPK + idx] = s;
    }
    // Value kernel matrices P[c][m=t*8+i][p], packed: Ppk[c][pp][m][e], p=2*pp+e
    //   t=0: A0, t=1,2: A1[o=t-1], t=3..5: A2[o=t-3]
    for (int idx = t0; idx < 6144; idx += stride) {
        int e = idx & 1, g = idx >> 1;
        int m = g % 48, pp = (g / 48) & 3, c = g / 192;
        int p = pp*2 + e, t = m >> 3, i = m & 7;
        float s = 0.f;
        if (t == 0) {
            for (int b = 0; b < NB; b++) s += vp0[c*NB + b] * vb0[b*64 + i*8 + p];
        } else if (t <= 2) {
            int o = t - 1;
            for (int b = 0; b < NB; b++) s += vp1[c*NB + b] * vb1[(b*2 + o)*64 + i*8 + p];
        } else {
            int o = t - 3;
            for (int b = 0; b < NB; b++) s += vp2[c*NB + b] * vb2[(b*3 + o)*64 + i*8 + p];
        }
        ws[WS_PPK + idx] = s;
    }
}

// ---------------------------------------------------------------------------
// Main kernel: one wave32 per vertex, f32 WMMA 16x16x4 for all contractions.
// ---------------------------------------------------------------------------
__global__ __launch_bounds__(WAVES * 32)
void attn_kernel(const float* __restrict__ x,          // [V][C*N]
                 const int* __restrict__ nbr,          // [V][MAXN]
                 const unsigned char* __restrict__ msk,// [V][MAXN] (bool)
                 const float* __restrict__ ptm,        // [V][MAXN][8][8]
                 const float* __restrict__ relpos,     // [V][MAXN][2]
                 const float* __restrict__ tab,        // workspace tables
                 float* __restrict__ out)              // [V][C][N]
{
    __shared__ float s_tab[WS_TOTAL];
    __shared__ float s_fp[WAVES * MAXN * CH * GN];     // f'[n][c][p] per wave

    const int tid = threadIdx.x;

    // ---- Async staging of weight tables: global -> LDS without VGPR bounce.
    // GLOBAL_LOAD_ASYNC_TO_LDS_B128, GVS mode (SGPR base + 32-bit VGPR offset),
    // tracked by ASYNCcnt.
    {
        const unsigned nfetch = WS_TOTAL / 4;          // 2304 x b128
        for (unsigned j = tid; j < nfetch; j += blockDim.x) {
            const unsigned fidx = j * 4u;
            const unsigned ldsa = (unsigned)(uintptr_t)&s_tab[fidx];
            const unsigned goff = fidx * 4u;
            asm volatile("global_load_async_to_lds_b128 %0, %1, %2"
                         :: "v"(ldsa), "v"(goff), "s"(tab)
                         : "memory");
        }
        asm volatile("s_wait_asynccnt 0x0" ::: "memory");
    }

    const int lane = tid & 31;
    const int w    = tid >> 5;
    const int li   = lane & 15;
    const int hi   = lane >> 4;
    const int v    = blockIdx.x * WAVES + w;
    float* FP = &s_fp[w * (MAXN * CH * GN)];

    // ---- Prefetch the random 512B neighbor-feature gathers + ptm stream
    if (lane < MAXN) {
        const int pn = nbr[v * MAXN + lane];
        __builtin_prefetch(x + (size_t)pn * 128, 0, 1);
        __builtin_prefetch(ptm + ((size_t)v * MAXN + lane) * 64, 0, 1);
    }
    __syncthreads();   // tables visible block-wide

    // ---- K[i] = sum_{c,p} W_K[c][i][p] * x[v][c*8+p]  (lane-parallel + reduce)
    const float4 xv = *(const float4*)(x + (size_t)v * 128 + lane * 4);
    float K[8];
#pragma unroll
    for (int i = 0; i < 8; i++) K[i] = 0.f;
#pragma unroll
    for (int k = 0; k < 4; k++) {
        const int cp = lane * 4 + k;
        const float xs = (&xv.x)[k];
        const float4* wk = (const float4*)&s_tab[WS_WKT + cp * 8];
        const float4 w0 = wk[0], w1 = wk[1];
#pragma unroll
        for (int i = 0; i < 4; i++) K[i]     += xs * (&w0.x)[i];
#pragma unroll
        for (int i = 0; i < 4; i++) K[i + 4] += xs * (&w1.x)[i];
    }
#pragma unroll
    for (int m = 1; m < 32; m <<= 1)
#pragma unroll
        for (int i = 0; i < 8; i++) K[i] += __shfl_xor(K[i], m, 32);

    // ---- Stage 1: f'[n] = Xg_n * R_nT  via 2x WMMA f32 16x16x4 per neighbor
    for (int n = 0; n < MAXN; n++) {
        const int nb = nbr[v * MAXN + n];
        const float mf = msk[v * MAXN + n] ? 1.f : 0.f;
        const float* xn = x + (size_t)nb * 128;
        // A operand (16x4, rows = channel): lane li holds Xg[li][2hi + {0,1}]
        v2f a0 = *(const v2f*)(xn + li * 8 + hi * 2);
        v2f a1 = *(const v2f*)(xn + li * 8 + 4 + hi * 2);
        a0 *= mf;  a1 *= mf;
        // B operand (4x16): B[k][i] = R[i][k]. Lanes li>=8 feed dead D columns,
        // so let them load duplicate rows (no EXEC divergence, no branches).
        const float* R = ptm + ((size_t)v * MAXN + n) * 64;
        v2f b0 = *(const v2f*)(R + (li & 7) * 8 + hi * 2);
        v2f b1 = *(const v2f*)(R + (li & 7) * 8 + 4 + hi * 2);
        v8f f = {};
        f = __builtin_amdgcn_wmma_f32_16x16x4_f32(false, a0, false, b0, (short)0, f, false, false);
        f = __builtin_amdgcn_wmma_f32_16x16x4_f32(false, a1, false, b1, (short)0, f, false, false);
        // D[m=c, ncol=p]: VGPR r, lane(li,hi) -> f'[c = r+8*hi][p = li] (li<8 valid)
        if (li < 8) {
#pragma unroll
            for (int r = 0; r < 8; r++)
                FP[(n * CH + hi * 8 + r) * GN + li] = f[r];
        }
    }

    // ---- Stage 2: Q[n][i] = sum_{cp} FP[n][cp] * W_Q[cp][i], K-dim=128 via 32 chunks
    v8f q = {};
    for (int qk = 0; qk < 32; qk++) {
        v2f a = *(const v2f*)(&FP[li * 128 + qk * 4 + hi * 2]);
        v2f b = *(const v2f*)(&s_tab[WS_WQPK + 2 * ((qk * 2 + hi) * 16 + li)]);
        q = __builtin_amdgcn_wmma_f32_16x16x4_f32(false, a, false, b, (short)0, q, false, false);
    }

    // ---- Attention: score[n] = mean_i relu(Q[n][i]+K[i]); normalize
    float att[8], u0r[8], u1r[8];
    float denom = 0.f;
#pragma unroll
    for (int r = 0; r < 8; r++) {
        const int n = r + 8 * hi;
        float t = 0.f;
        if (li < 8) {
            const float z = q[r] + K[li];
            t = z > 0.f ? z : 0.f;
        }
        t += __shfl_xor(t, 1, 32);
        t += __shfl_xor(t, 2, 32);
        t += __shfl_xor(t, 4, 32);
        t += __shfl_xor(t, 8, 32);
        float sc = t * 0.125f;
        sc *= msk[v * MAXN + n] ? 1.f : 0.f;
        att[r] = sc;
        denom += sc;
        const v2f u = *(const v2f*)(relpos + ((size_t)v * MAXN + n) * 2);
        u0r[r] = u[0];  u1r[r] = u[1];
    }
    denom += __shfl_xor(denom, 16, 32);
    denom = fmaxf(denom, 1e-8f);
    const float inv = 1.f / denom;
#pragma unroll
    for (int r = 0; r < 8; r++) att[r] *= inv;

    // ---- Stage 3: per channel, g[n][m] = sum_p f'[n][c][p] * P[c][m][p]
    //      (M=16 neighbors, N=48 -> 3 tiles, K=8 -> 2 chunks; 100% dense)
    for (int c = 0; c < CH; c++) {
        v2f a0 = *(const v2f*)(&FP[(li * CH + c) * GN + hi * 2]);
        v2f a1 = *(const v2f*)(&FP[(li * CH + c) * GN + 4 + hi * 2]);
        float pv[8];
#pragma unroll
        for (int r = 0; r < 8; r++) pv[r] = 0.f;
#pragma unroll
        for (int T = 0; T < 3; T++) {
            v8f acc = {};
            v2f b0 = *(const v2f*)(&s_tab[WS_PPK + 2 * ((c * 4 + 0 + hi) * 48 + T * 16 + li)]);
            acc = __builtin_amdgcn_wmma_f32_16x16x4_f32(false, a0, false, b0, (short)0, acc, false, false);
            v2f b1 = *(const v2f*)(&s_tab[WS_PPK + 2 * ((c * 4 + 2 + hi) * 48 + T * 16 + li)]);
            acc = __builtin_amdgcn_wmma_f32_16x16x4_f32(false, a1, false, b1, (short)0, acc, false, false);
            // lane owns (t = 2T + (li>=8), i = li&7); apply Taylor coefficient s_t[n]
#pragma unroll
            for (int r = 0; r < 8; r++) {
                const float u0 = u0r[r], u1 = u1r[r];
                float s;
                if (T == 0)      s = (li < 8) ? 1.f            : u0;
                else if (T == 1) s = (li < 8) ? u1             : u0 * u0;
                else             s = (li < 8) ? 2.f * u0 * u1  : u1 * u1;
                pv[r] += s * acc[r];
            }
        }
        float o = 0.f;
#pragma unroll
        for (int r = 0; r < 8; r++) {
            pv[r] += __shfl_xor(pv[r], 8, 32);   // fold t-parity partner
            o += att[r] * pv[r];                 // weight by attention, sum 8 neighbors
        }
        o += __shfl_xor(o, 16, 32);              // fold other neighbor half
        if (lane < 8)
            out[(size_t)v * 128 + c * 8 + lane] = o;
    }
}

// ---------------------------------------------------------------------------
extern "C" void kernel_launch(void* const* d_in, const int* in_sizes, int n_in,
                              void* d_out, int out_size, void* d_ws, size_t ws_size,
                              hipStream_t stream) {
    const float*         x      = (const float*)d_in[0];
    const int*           nbrs   = (const int*)d_in[1];
    const unsigned char* mask   = (const unsigned char*)d_in[2];
    const float*         ptm    = (const float*)d_in[3];
    const float*         relpos = (const float*)d_in[4];
    const float*         basis  = (const float*)d_in[5];
    const float*         qc     = (const float*)d_in[6];
    const float*         kc     = (const float*)d_in[7];
    const float*         vb0    = (const float*)d_in[8];
    const float*         vb1    = (const float*)d_in[9];
    const float*         vb2    = (const float*)d_in[10];
    const float*         vp0    = (const float*)d_in[11];
    const float*         vp1    = (const float*)d_in[12];
    const float*         vp2    = (const float*)d_in[13];
    float* ws  = (float*)d_ws;
    float* out = (float*)d_out;

    prep_kernel<<<36, 256, 0, stream>>>(basis, qc, kc, vb0, vb1, vb2, vp0, vp1, vp2, ws);
    attn_kernel<<<NV / WAVES, WAVES * 32, 0, stream>>>(x, nbrs, mask, ptm, relpos, ws, out);
}